// MultiInfAffine_36352603193791
// MI455X (gfx1250) — compile-verified
//
#include <hip/hip_runtime.h>
#include <hip/hip_bf16.h>
#include <math.h>

typedef __attribute__((ext_vector_type(2))) float v2f;
typedef __attribute__((ext_vector_type(8))) float v8f;

#define NPTS 250000
#define DDIM 16
#define LLAY 6
#define KCMP 64

// ---------------------------------------------------------------------------
// Prep: normalize mu columns over D and transpose to [L][K][D]; wv = exp(-w^2)
// ---------------------------------------------------------------------------
__global__ void MultiInfAffine_prep(const float* __restrict__ mus,   // [L,D,K]
                                    const float* __restrict__ ws,    // [L]
                                    float* __restrict__ mu_t,        // [L,K,D]
                                    float* __restrict__ wv) {        // [L]
    int tid = blockIdx.x * blockDim.x + threadIdx.x;
    if (tid < LLAY * KCMP) {
        int l = tid >> 6;
        int k = tid & 63;
        const float* p = mus + (size_t)l * DDIM * KCMP + k;
        float tmp[DDIM];
        float ss = 0.f;
#pragma unroll
        for (int d = 0; d < DDIM; ++d) {
            float v = p[(size_t)d * KCMP];
            tmp[d] = v;
            ss += v * v;
        }
        float inv = rsqrtf(ss);
        float* q = mu_t + ((size_t)l * KCMP + k) * DDIM;
#pragma unroll
        for (int d = 0; d < DDIM; ++d) q[d] = tmp[d] * inv;
    }
    if (tid < LLAY) {
        float w = ws[tid];
        wv[tid] = __expf(-w * w);
    }
}

// ---------------------------------------------------------------------------
// Main: one wave32 handles 16 points.
//   C[comp, point] tiles via V_WMMA_F32_16X16X4_F32, chained over D in 4-chunks.
//   Lane layout (16x16 f32 C/D): n = lane%16 (point), rows m = vgpr + 8*(lane/16).
//   => lane (n, h) holds comps {t*16 + 8h + r : r=0..7, t=0..3}  (32 of 64).
//   Soft-min over K: in-register LSE (32 vals) + one shfl_xor(16) merge.
// ---------------------------------------------------------------------------
__global__ void __launch_bounds__(256)
MultiInfAffine_main(const float* __restrict__ xs,      // [N,16]
                    const float* __restrict__ alphas,  // [L,64]
                    const float* __restrict__ mu_t,    // [L,64,16] normalized
                    const float* __restrict__ wv,      // [L]
                    float* __restrict__ out, int nPts) {
    __shared__ __align__(16) float smu[LLAY * KCMP * DDIM];   // 24 KB
    __shared__ __align__(16) float salpha[LLAY * KCMP];       // 1.5 KB

    const int tid  = threadIdx.x;
    // cooperative stage of parameters into LDS
    for (int i = tid; i < LLAY * KCMP * DDIM; i += 256) smu[i] = mu_t[i];
    for (int i = tid; i < LLAY * KCMP; i += 256) salpha[i] = alphas[i];
    __syncthreads();

    const int wave = tid >> 5;
    const int lane = tid & 31;
    const int n    = lane & 15;   // point within tile / N column
    const int h    = lane >> 4;   // half-wave
    const long base = ((long)blockIdx.x * 8 + wave) * 16;
    if (base >= nPts) return;     // wave-uniform: EXEC stays all-ones for WMMA

    // B operands: xs tile, 4 x (K=4) dim-chunks. Lane holds dims d0=4*kk+2h, +1.
    v2f b[4];
#pragma unroll
    for (int kk = 0; kk < 4; ++kk) {
        const float* p = xs + (base + n) * DDIM + 4 * kk + 2 * h;
        b[kk] = *(const v2f*)p;
    }

    float F = 0.f;
#pragma unroll
    for (int l = 0; l < LLAY; ++l) {
        const float* muL = smu + l * KCMP * DDIM;
        v8f acc[4];
#pragma unroll
        for (int t = 0; t < 4; ++t) acc[t] = (v8f){};

#pragma unroll
        for (int kk = 0; kk < 4; ++kk) {
            const int d0 = 4 * kk + 2 * h;
#pragma unroll
            for (int t = 0; t < 4; ++t) {
                // A operand: comp c = t*16 + n holds dims (d0, d0+1)
                v2f a = *(const v2f*)(muL + (t * 16 + n) * DDIM + d0);
                acc[t] = __builtin_amdgcn_wmma_f32_16x16x4_f32(
                    false, a, false, b[kk], (short)0, acc[t], false, false);
            }
        }

        // per-lane soft-min contribution over its 32 components
        float vals[32];
        float vmax = -3.0e38f;
#pragma unroll
        for (int t = 0; t < 4; ++t) {
#pragma unroll
            for (int r = 0; r < 8; ++r) {
                const int c = t * 16 + 8 * h + r;
                float x = acc[t][r];
                x = fminf(fmaxf(x, -1.f + 1e-7f), 1.f - 1e-7f);
                float d = acosf(x);
                float cost = 0.5f * d * d + salpha[l * KCMP + c];
                float v = -10.0f * cost;          // -cost / gamma, gamma = 0.1
                vals[t * 8 + r] = v;
                vmax = fmaxf(vmax, v);
            }
        }
        float s = 0.f;
#pragma unroll
        for (int i = 0; i < 32; ++i) s += __expf(vals[i] - vmax);

        // merge the two half-wave partials for this point (comps split at 16)
        float m2 = __shfl_xor(vmax, 16, 32);
        float s2 = __shfl_xor(s, 16, 32);
        float M  = fmaxf(vmax, m2);
        float st = s * __expf(vmax - M) + s2 * __expf(m2 - M);
        float mincost = 0.1f * (M + __logf(st));

        float wvl = wv[l];
        F = wvl * fmaxf(F, 0.f) + (1.f - wvl) * mincost;
    }

    if (h == 0) {
        // 0.1 * LSE(-F/0.1, 0), numerically stable
        float a = -10.0f * F;
        float r = 0.1f * (fmaxf(a, 0.f) + log1pf(__expf(-fabsf(a))));
        out[base + n] = r;
    }
}

// ---------------------------------------------------------------------------
extern "C" void kernel_launch(void* const* d_in, const int* in_sizes, int n_in,
                              void* d_out, int out_size, void* d_ws, size_t ws_size,
                              hipStream_t stream) {
    const float* xs     = (const float*)d_in[0];
    const float* mus    = (const float*)d_in[1];
    const float* alphas = (const float*)d_in[2];
    const float* ws     = (const float*)d_in[3];
    float* out = (float*)d_out;

    float* mu_t = (float*)d_ws;                    // 6*64*16 floats = 24 KB
    float* wv   = mu_t + LLAY * KCMP * DDIM;       // 6 floats

    const int nPts = in_sizes[0] / DDIM;           // 250000

    MultiInfAffine_prep<<<1, 384, 0, stream>>>(mus, ws, mu_t, wv);

    const int ptsPerBlock = 128;                   // 8 waves * 16 points
    const int grid = (nPts + ptsPerBlock - 1) / ptsPerBlock;
    MultiInfAffine_main<<<grid, 256, 0, stream>>>(xs, alphas, mu_t, wv, out, nPts);
}